// GCN_12876311953783
// MI455X (gfx1250) — compile-verified
//
#include <hip/hip_runtime.h>

typedef __attribute__((ext_vector_type(2))) float v2f;
typedef __attribute__((ext_vector_type(8))) float v8f;

#define IN_D  32
#define HID_D 32
#define OUT_D 16

// ---------------------------------------------------------------------------
// Utility: zero a float buffer, float4-vectorized (b128 stores).
// ---------------------------------------------------------------------------
__global__ void gcn_zero_kernel(float* __restrict__ p, long long n) {
    long long i = ((long long)blockIdx.x * blockDim.x + threadIdx.x) * 4;
    if (i + 3 < n) {
        *(float4*)(p + i) = make_float4(0.f, 0.f, 0.f, 0.f);
    } else {
        for (long long j = i; j < n; ++j) p[j] = 0.0f;
    }
}

// ---------------------------------------------------------------------------
// Degree accumulation: deg_out[src[e]] += 1 ; deg_in[dst[e]] += 1
// ---------------------------------------------------------------------------
__global__ void gcn_degree_kernel(const int* __restrict__ src,
                                  const int* __restrict__ dst,
                                  float* __restrict__ degO,
                                  float* __restrict__ degI, int E) {
    int e = blockIdx.x * blockDim.x + threadIdx.x;
    if (e < E) {
        atomicAdd(&degO[src[e]], 1.0f);
        atomicAdd(&degI[dst[e]], 1.0f);
    }
}

// ---------------------------------------------------------------------------
// norm = clip(deg, 1)^-0.5
// ---------------------------------------------------------------------------
__global__ void gcn_norm_kernel(const float* __restrict__ degO,
                                const float* __restrict__ degI,
                                float* __restrict__ ns,
                                float* __restrict__ nd, int n) {
    int i = blockIdx.x * blockDim.x + threadIdx.x;
    if (i < n) {
        ns[i] = rsqrtf(fmaxf(degO[i], 1.0f));
        nd[i] = rsqrtf(fmaxf(degI[i], 1.0f));
    }
}

// ---------------------------------------------------------------------------
// GEMM1 (WMMA f32 16x16x4): h1 = (features * norm_src[:,None]) @ W1
// One wave per 16-row tile; two 16-wide N tiles (HID = 32); K = 32 in 8 steps.
// A layout (16x4 f32): lanes 0-15 -> M=lane, K={0,1}; lanes 16-31 -> K={2,3}.
// B layout (4x16 f32): lanes 0-15 -> N=lane, K={0,1}; lanes 16-31 -> K={2,3}.
// C/D layout: VGPR j, lanes 0-15 -> (M=j, N=lane); lanes 16-31 -> (M=j+8).
// ---------------------------------------------------------------------------
__global__ void gcn_gemm1_kernel(const float* __restrict__ feat,
                                 const float* __restrict__ ns,
                                 const float* __restrict__ W1,
                                 float* __restrict__ h1, int n) {
    const int wave = threadIdx.x >> 5;
    const int lane = threadIdx.x & 31;
    const int tile = blockIdx.x * (blockDim.x >> 5) + wave;
    const int row0 = tile * 16;
    if (row0 >= n) return;  // wave-uniform: EXEC stays all-1s for WMMA

    const int half = lane >> 4;   // 0 or 1
    const int mrow = lane & 15;
    const int arow = row0 + mrow;
    const float scale = ns[arow];

    v8f acc0 = {};
    v8f acc1 = {};
#pragma unroll
    for (int k0 = 0; k0 < IN_D; k0 += 4) {
        const int kk = k0 + 2 * half;
        v2f a, b0, b1;
        const float* ap = feat + (long long)arow * IN_D + kk;
        a.x = ap[0] * scale;
        a.y = ap[1] * scale;
        b0.x = W1[kk * HID_D + mrow];
        b0.y = W1[(kk + 1) * HID_D + mrow];
        b1.x = W1[kk * HID_D + 16 + mrow];
        b1.y = W1[(kk + 1) * HID_D + 16 + mrow];
        acc0 = __builtin_amdgcn_wmma_f32_16x16x4_f32(false, a, false, b0,
                                                     (short)0, acc0, false, false);
        acc1 = __builtin_amdgcn_wmma_f32_16x16x4_f32(false, a, false, b1,
                                                     (short)0, acc1, false, false);
    }
#pragma unroll
    for (int j = 0; j < 8; ++j) {
        const int m = row0 + j + 8 * half;
        h1[(long long)m * HID_D + mrow]      = acc0[j];
        h1[(long long)m * HID_D + 16 + mrow] = acc1[j];
    }
}

// ---------------------------------------------------------------------------
// GEMM2 (WMMA f32 16x16x4), fusing layer-1 epilogue into the A load:
//   x1    = relu(agg1 * norm_dst[:,None] + b1)
//   h2    = (x1 * norm_src[:,None]) @ W2            (HID=32 -> OUT=16)
// ---------------------------------------------------------------------------
__global__ void gcn_gemm2_kernel(const float* __restrict__ agg1,
                                 const float* __restrict__ ns,
                                 const float* __restrict__ nd,
                                 const float* __restrict__ b1,
                                 const float* __restrict__ W2,
                                 float* __restrict__ h2, int n) {
    const int wave = threadIdx.x >> 5;
    const int lane = threadIdx.x & 31;
    const int tile = blockIdx.x * (blockDim.x >> 5) + wave;
    const int row0 = tile * 16;
    if (row0 >= n) return;  // wave-uniform

    const int half = lane >> 4;
    const int mrow = lane & 15;
    const int arow = row0 + mrow;
    const float s_ns = ns[arow];
    const float s_nd = nd[arow];

    v8f acc = {};
#pragma unroll
    for (int k0 = 0; k0 < HID_D; k0 += 4) {
        const int kk = k0 + 2 * half;
        v2f a, b;
        const float* ap = agg1 + (long long)arow * HID_D + kk;
        a.x = fmaxf(ap[0] * s_nd + b1[kk],     0.0f) * s_ns;
        a.y = fmaxf(ap[1] * s_nd + b1[kk + 1], 0.0f) * s_ns;
        b.x = W2[kk * OUT_D + mrow];
        b.y = W2[(kk + 1) * OUT_D + mrow];
        acc = __builtin_amdgcn_wmma_f32_16x16x4_f32(false, a, false, b,
                                                    (short)0, acc, false, false);
    }
#pragma unroll
    for (int j = 0; j < 8; ++j) {
        const int m = row0 + j + 8 * half;
        h2[(long long)m * OUT_D + mrow] = acc[j];
    }
}

// ---------------------------------------------------------------------------
// Edge scatter: one thread per (edge, float4 column group).
//   agg[dst[e]*D + g*4 .. +3] += h[src[e]*D + g*4 .. +3] * w[e]
// Gather side uses global_load_b128; scatter side is 4 scalar f32 atomics
// (no packed f32 atomic exists). Index/weight loads amortized 4x.
// ---------------------------------------------------------------------------
template <int D>
__global__ void gcn_scatter_kernel(const float* __restrict__ h,
                                   const int* __restrict__ src,
                                   const int* __restrict__ dst,
                                   const float* __restrict__ w,
                                   float* __restrict__ agg, long long total) {
    constexpr int G = D / 4;  // float4 groups per edge
    long long t = (long long)blockIdx.x * blockDim.x + threadIdx.x;
    if (t >= total) return;
    const int e = (int)(t / G);
    const int g = (int)(t % G);
    const int s = src[e];
    const int d = dst[e];
    const float wt = w[e];
    const float4 hv = *(const float4*)(h + (long long)s * D + g * 4);
    float* ap = agg + (long long)d * D + g * 4;
    atomicAdd(ap + 0, hv.x * wt);
    atomicAdd(ap + 1, hv.y * wt);
    atomicAdd(ap + 2, hv.z * wt);
    atomicAdd(ap + 3, hv.w * wt);
}

// ---------------------------------------------------------------------------
// Final epilogue (in place on d_out, float4-vectorized):
//   out = out * norm_dst[:,None] + b2
// ---------------------------------------------------------------------------
__global__ void gcn_finalize_kernel(float* __restrict__ out,
                                    const float* __restrict__ nd,
                                    const float* __restrict__ b2, long long total4) {
    constexpr int G = OUT_D / 4;  // 4 float4 groups per row
    long long t = (long long)blockIdx.x * blockDim.x + threadIdx.x;
    if (t >= total4) return;
    const int i = (int)(t / G);
    const int g = (int)(t % G);
    const float s = nd[i];
    const float4 bv = *(const float4*)(b2 + g * 4);
    float4* op = (float4*)(out + (long long)i * OUT_D + g * 4);
    float4 v = *op;
    v.x = v.x * s + bv.x;
    v.y = v.y * s + bv.y;
    v.z = v.z * s + bv.z;
    v.w = v.w * s + bv.w;
    *op = v;
}

// ---------------------------------------------------------------------------
// Launcher
// ---------------------------------------------------------------------------
extern "C" void kernel_launch(void* const* d_in, const int* in_sizes, int n_in,
                              void* d_out, int out_size, void* d_ws, size_t ws_size,
                              hipStream_t stream) {
    const float* feat = (const float*)d_in[0];
    const int*   src  = (const int*)d_in[1];
    const int*   dst  = (const int*)d_in[2];
    const float* ew   = (const float*)d_in[3];
    const float* W1   = (const float*)d_in[4];
    const float* b1   = (const float*)d_in[5];
    const float* W2   = (const float*)d_in[6];
    const float* b2   = (const float*)d_in[7];
    float* out = (float*)d_out;

    const int n = in_sizes[0] / IN_D;   // 100000
    const int E = in_sizes[1];          // 1600000

    // Workspace layout (floats)
    float* degO = (float*)d_ws;
    float* degI = degO + n;
    float* ns   = degI + n;
    float* nd   = ns + n;
    float* h1   = nd + n;
    float* agg1 = h1 + (long long)n * HID_D;
    float* h2   = agg1 + (long long)n * HID_D;

    const int B = 256;
    auto blocks = [&](long long work) { return (unsigned)((work + B - 1) / B); };

    // 1) Zero accumulators (degO+degI contiguous) and d_out (b128 stores).
    gcn_zero_kernel<<<blocks((2LL * n + 3) / 4), B, 0, stream>>>(degO, 2LL * n);
    gcn_zero_kernel<<<blocks(((long long)n * HID_D + 3) / 4), B, 0, stream>>>(
        agg1, (long long)n * HID_D);
    gcn_zero_kernel<<<blocks(((long long)out_size + 3) / 4), B, 0, stream>>>(
        out, (long long)out_size);

    // 2) Degrees.
    gcn_degree_kernel<<<blocks(E), B, 0, stream>>>(src, dst, degO, degI, E);

    // 3) Norms.
    gcn_norm_kernel<<<blocks(n), B, 0, stream>>>(degO, degI, ns, nd, n);

    // 4) Layer-1 dense transform (WMMA f32).
    const int tiles = (n + 15) / 16;
    const int wavesPerBlock = B / 32;
    const unsigned gemmBlocks = (unsigned)((tiles + wavesPerBlock - 1) / wavesPerBlock);
    gcn_gemm1_kernel<<<gemmBlocks, B, 0, stream>>>(feat, ns, W1, h1, n);

    // 5) Layer-1 edge scatter-add (8 float4-groups per edge).
    gcn_scatter_kernel<HID_D><<<blocks((long long)E * (HID_D / 4)), B, 0, stream>>>(
        h1, src, dst, ew, agg1, (long long)E * (HID_D / 4));

    // 6) Layer-2 dense transform (fused relu/bias/norms, WMMA f32).
    gcn_gemm2_kernel<<<gemmBlocks, B, 0, stream>>>(agg1, ns, nd, b1, W2, h2, n);

    // 7) Layer-2 edge scatter-add directly into d_out (4 groups per edge).
    gcn_scatter_kernel<OUT_D><<<blocks((long long)E * (OUT_D / 4)), B, 0, stream>>>(
        h2, src, dst, ew, out, (long long)E * (OUT_D / 4));

    // 8) Final epilogue in place (float4).
    gcn_finalize_kernel<<<blocks((long long)n * (OUT_D / 4)), B, 0, stream>>>(
        out, nd, b2, (long long)n * (OUT_D / 4));
}